// GCNLayer_27436251086936
// MI455X (gfx1250) — compile-verified
//
#include <hip/hip_runtime.h>
#include <hip/hip_bf16.h>
#include <cstddef>

// ---------------------------------------------------------------------------
// GCN layer for MI455X (gfx1250):
//   support = X @ W            (fp32 WMMA 16x16x4, B staged via async->LDS)
//   out     = relu(segsum(vals * support[cols], rows))
// ---------------------------------------------------------------------------

#define N_NODES 100000
#define D_IN    256
#define D_OUT   128
#define K_CHUNK 64          // K rows of W staged per LDS round (32 KB)

typedef __attribute__((ext_vector_type(2))) float v2f;
typedef __attribute__((ext_vector_type(8))) float v8f;
typedef int v4i_g __attribute__((vector_size(16)));   // matches builtin param type

#define AS1 __attribute__((address_space(1)))
#define AS3 __attribute__((address_space(3)))

// ---------------------------------------------------------------------------
// Kernel 1: support = X @ W  via V_WMMA_F32_16X16X4_F32
//   block = 256 threads = 8 waves; each wave owns one 16x16 output tile.
//   Block covers 16 rows x 128 cols; grid.x = N_NODES/16 = 6250.
//   W is staged chunk-wise into LDS with GLOBAL_LOAD_ASYNC_TO_LDS_B128
//   (ASYNCcnt); B operands are read from LDS (conflict-free ds loads).
// ---------------------------------------------------------------------------
__global__ void gcn_gemm_wmma(const float* __restrict__ X,
                              const float* __restrict__ W,
                              float* __restrict__ S) {
    __shared__ float wlds[K_CHUNK * D_OUT];      // 32 KB

    const int lane  = threadIdx.x & 31;
    const int wave  = threadIdx.x >> 5;          // 0..7 -> 16-col tile
    const int mBase = blockIdx.x * 16;
    const int nBase = wave * 16;

    const int mRow  = lane & 15;                 // M = lane % 16 (both halves)
    const int kHalf = (lane >> 4) * 2;           // lanes 16..31 carry K+2,K+3
    const int nCol  = nBase + (lane & 15);

    const float* __restrict__ xrow = X + (size_t)(mBase + mRow) * D_IN;

    v8f acc = {};
    for (int kc = 0; kc < D_IN; kc += K_CHUNK) {
        // ---- async bulk copy: W[kc .. kc+K_CHUNK) x 128 -> LDS (32 KB) ----
        // 256 threads x 8 passes x 16 B = 32 KB; each lane copies its own
        // 16-byte slot. Builtin form keeps the LDS write visible to LLVM so
        // the ds-load consumers are not folded away.
        {
            float*       ldst = &wlds[threadIdx.x * 4];
            const float* gsrc = W + (size_t)kc * D_OUT + threadIdx.x * 4;
            #pragma unroll
            for (int p = 0; p < (K_CHUNK * D_OUT) / 1024; ++p) {   // 8 passes
                float*       l = ldst + p * 1024;
                const float* g = gsrc + p * 1024;
#if __has_builtin(__builtin_amdgcn_global_load_async_to_lds_b128)
                __builtin_amdgcn_global_load_async_to_lds_b128(
                    (AS1 v4i_g*)g, (AS3 v4i_g*)l, /*offset=*/0, /*cpol=*/0);
#else
                *(float4*)l = *(const float4*)g;   // synchronous fallback
#endif
            }
#if __has_builtin(__builtin_amdgcn_s_wait_asynccnt)
            __builtin_amdgcn_s_wait_asynccnt(0);
#else
            asm volatile("s_wait_asynccnt 0x0" ::: "memory");
#endif
        }
        __syncthreads();

        // ---- 16 WMMAs over this K chunk; B pairs served from LDS ----
        #pragma unroll 4
        for (int kk = 0; kk < K_CHUNK; kk += 4) {
            // A (16x4 f32): lane holds X[M][k+kHalf], X[M][k+kHalf+1]
            v2f a = *(const v2f*)(xrow + kc + kk + kHalf);
            // B (4x16 f32): VGPR0 = row K (lanes0-15) / K+2 (lanes16-31)
            const float* wl = &wlds[(kk + kHalf) * D_OUT + nCol];
            v2f b;
            b.x = wl[0];
            b.y = wl[D_OUT];
            acc = __builtin_amdgcn_wmma_f32_16x16x4_f32(
                      /*neg_a=*/false, a, /*neg_b=*/false, b,
                      /*c_mod=*/(short)0, acc, /*reuse_a=*/false, /*reuse_b=*/false);
        }
        __syncthreads();    // all waves done reading before next chunk overwrite
    }

    // C/D layout: VGPR r -> row (mBase + r + 8*(lane>=16)), col nCol
    const int rowOff = (lane >> 4) * 8;
    #pragma unroll
    for (int r = 0; r < 8; ++r) {
        S[(size_t)(mBase + rowOff + r) * D_OUT + nCol] = acc[r];
    }
}

// ---------------------------------------------------------------------------
// Kernel 2: SpMM scatter-add. One wave per edge; lane handles 4 contiguous
// floats of the 128-wide row (float4 load -> global_load_b128; 4x HW f32
// atomics -> global_atomic_add_f32). support/out live in L2 (51 MB each).
// ---------------------------------------------------------------------------
__global__ void gcn_spmm_scatter(const float* __restrict__ support,
                                 const int*   __restrict__ rows,
                                 const int*   __restrict__ cols,
                                 const float* __restrict__ vals,
                                 float* __restrict__ out,
                                 int nEdges) {
    const long long gtid = (long long)blockIdx.x * blockDim.x + threadIdx.x;
    const int e    = (int)(gtid >> 5);
    const int lane = (int)(gtid & 31);
    if (e >= nEdges) return;

    const int   c = cols[e];
    const int   r = rows[e];
    const float v = vals[e];

    // Prefetch the gather row of an edge ~8 ahead (global_prefetch_b8).
    if (lane == 0 && e + 8 < nEdges) {
        __builtin_prefetch((const void*)(support + (size_t)cols[e + 8] * D_OUT), 0, 0);
    }

    const float4 s = *(const float4*)(support + (size_t)c * D_OUT + lane * 4);
    float* dst = out + (size_t)r * D_OUT + lane * 4;

    unsafeAtomicAdd(dst + 0, v * s.x);
    unsafeAtomicAdd(dst + 1, v * s.y);
    unsafeAtomicAdd(dst + 2, v * s.z);
    unsafeAtomicAdd(dst + 3, v * s.w);
}

// ---------------------------------------------------------------------------
// Kernel 3: in-place fused ReLU over the accumulated output.
// ---------------------------------------------------------------------------
__global__ void gcn_relu(float* __restrict__ out, int n4) {
    const int i = blockIdx.x * blockDim.x + threadIdx.x;
    if (i >= n4) return;
    float4* p = (float4*)out + i;
    float4 x = *p;
    x.x = fmaxf(x.x, 0.f);
    x.y = fmaxf(x.y, 0.f);
    x.z = fmaxf(x.z, 0.f);
    x.w = fmaxf(x.w, 0.f);
    *p = x;
}

// ---------------------------------------------------------------------------
extern "C" void kernel_launch(void* const* d_in, const int* in_sizes, int n_in,
                              void* d_out, int out_size, void* d_ws, size_t ws_size,
                              hipStream_t stream) {
    const float* features  = (const float*)d_in[0];   // [N_NODES, D_IN]
    const float* weight    = (const float*)d_in[1];   // [D_IN, D_OUT]
    const int*   edge_rows = (const int*)  d_in[2];   // [E]
    const int*   edge_cols = (const int*)  d_in[3];   // [E]
    const float* edge_vals = (const float*)d_in[4];   // [E]
    const int    nEdges    = in_sizes[2];

    float* out     = (float*)d_out;                   // [N_NODES, D_OUT]
    float* support = (float*)d_ws;                    // [N_NODES, D_OUT] scratch

    // 1) support = X @ W (WMMA f32, async-LDS staged B)
    gcn_gemm_wmma<<<N_NODES / 16, 256, 0, stream>>>(features, weight, support);

    // 2) zero accumulator (graph-capture-safe async memset)
    (void)hipMemsetAsync(d_out, 0, (size_t)out_size * sizeof(float), stream);

    // 3) scatter-add: 1 wave per edge, 256-thread blocks = 8 edges/block
    {
        const long long threads = (long long)nEdges * 32;
        const int blocks = (int)((threads + 255) / 256);
        gcn_spmm_scatter<<<blocks, 256, 0, stream>>>(support, edge_rows, edge_cols,
                                                     edge_vals, out, nEdges);
    }

    // 4) fused ReLU (in place)
    {
        const int n4 = out_size / 4;
        gcn_relu<<<(n4 + 255) / 256, 256, 0, stream>>>(out, n4);
    }
}